// FMoE_33767032881217
// MI455X (gfx1250) — compile-verified
//
#include <hip/hip_runtime.h>
#include <stdint.h>

// FMoE routed GEMM for gfx1250 (MI455X, wave32, WMMA + async global->LDS).
// out[t] = weight[gate[t]] @ inp[t], fp32.
// Bucket tokens by expert into perm[], then per-expert dense GEMM using
// V_WMMA_F32_16X16X4_F32 with double-buffered LDS tiles filled by
// GLOBAL_LOAD_ASYNC_TO_LDS_B128 (ASYNCcnt-tracked).

typedef __attribute__((ext_vector_type(2))) float v2f;
typedef __attribute__((ext_vector_type(8))) float v8f;

#define NUM_T   8192
#define NUM_E   32
#define DIM_IN  1024
#define DIM_OUT 1024

#define NT 64                 // N columns per block
#define MT 128                // M tokens per chunk (8 waves x 16 rows)
#define KT 32                 // K chunk staged in LDS (double-buffered)
#define KTS (KT + 4)          // pad -> conflict-free b64 fragment reads

// Async global->LDS (gfx1250): VDST = LDS byte address, VADDR = 64-bit global
// address (GV mode). Tracked with ASYNCcnt.
#define ASYNC_LOAD_B128(ldsoff, gptr)                                        \
    asm volatile("global_load_async_to_lds_b128 %0, %1, off"                 \
                 :: "v"(ldsoff),                                             \
                    "v"((unsigned long long)(uintptr_t)(gptr))               \
                 : "memory")

#define WAIT_ASYNC0() asm volatile("s_wait_asynccnt 0x0" ::: "memory")

// ---------------- bucketing kernels ----------------

__global__ void fmoe_zero_ws(int* __restrict__ ws) {
    if (threadIdx.x < 96) ws[threadIdx.x] = 0;   // counts | offsets | cursor
}

__global__ void fmoe_histo(const int* __restrict__ gate, int* __restrict__ counts) {
    int t = blockIdx.x * blockDim.x + threadIdx.x;
    if (t < NUM_T) atomicAdd(&counts[gate[t]], 1);
}

__global__ void fmoe_scan(const int* __restrict__ counts,
                          int* __restrict__ offsets, int* __restrict__ cursor) {
    if (threadIdx.x == 0) {
        int s = 0;
        for (int e = 0; e < NUM_E; ++e) {
            offsets[e] = s;
            cursor[e]  = s;
            s += counts[e];
        }
    }
}

__global__ void fmoe_scatter(const int* __restrict__ gate, int* __restrict__ cursor,
                             int* __restrict__ perm) {
    int t = blockIdx.x * blockDim.x + threadIdx.x;
    if (t < NUM_T) {
        int p = atomicAdd(&cursor[gate[t]], 1);
        perm[p] = t;
    }
}

// ---------------- grouped GEMM kernel ----------------
// grid = (DIM_OUT/NT, NUM_E), block = 256 (8 waves).
// Each block: expert e, columns [n0, n0+64). Loops over 128-token M chunks;
// K staged in 32-wide double-buffered LDS chunks via async loads.

__global__ void __launch_bounds__(256)
fmoe_gemm(const float* __restrict__ inp,      // [T, IN]
          const float* __restrict__ weight,   // [E, OUT, IN]
          const int*   __restrict__ perm,     // [T] tokens grouped by expert
          const int*   __restrict__ counts,   // [E]
          const int*   __restrict__ offsets,  // [E] exclusive prefix
          float*       __restrict__ out)      // [T, OUT]
{
    const int e    = blockIdx.y;
    const int n0   = blockIdx.x * NT;
    const int cnt  = counts[e];
    if (cnt == 0) return;                      // block-uniform exit
    const int base = offsets[e];

    __shared__ float Xs[2][MT][KTS];           // 2*128*36*4 = 36864 B
    __shared__ float Ws[2][NT][KTS];           // 2* 64*36*4 = 18432 B

    const int tid  = threadIdx.x;
    const int lane = tid & 31;
    const int wave = tid >> 5;                 // 0..7 -> M sub-tile
    const int l16  = lane & 15;
    const int hi   = (lane >> 4) << 1;         // 0 for lanes 0-15, 2 for 16-31

    // per-thread staging coordinates: one float4 column, rows strided by 32
    const int rS = tid >> 3;                   // 0..31
    const int cS = (tid & 7) * 4;              // float4 column (floats)

    const float* wbase = weight + (size_t)e * DIM_OUT * DIM_IN + (size_t)n0 * DIM_IN;

    for (int m0 = 0; m0 < cnt; m0 += MT) {
        // cache this thread's 4 X-row tokens for the whole K loop
        int  tokr[4];
        bool valr[4];
        #pragma unroll
        for (int j = 0; j < 4; ++j) {
            const int m = m0 + rS + 32 * j;
            valr[j] = (m < cnt);
            tokr[j] = valr[j] ? perm[base + m] : 0;
        }

        // issue one K-chunk of async loads into buffer `buf`
        auto issue = [&](int buf, int k0) {
            #pragma unroll
            for (int i = 0; i < 2; ++i) {      // W: 64 rows x 32 floats
                const int r = rS + 32 * i;
                const unsigned lds = (unsigned)(uintptr_t)&Ws[buf][r][cS];
                ASYNC_LOAD_B128(lds, wbase + (size_t)r * DIM_IN + k0 + cS);
            }
            #pragma unroll
            for (int j = 0; j < 4; ++j) {      // X: 128 rows x 32 floats
                const int r = rS + 32 * j;
                if (valr[j]) {
                    const unsigned lds = (unsigned)(uintptr_t)&Xs[buf][r][cS];
                    ASYNC_LOAD_B128(lds, inp + (size_t)tokr[j] * DIM_IN + k0 + cS);
                } else {
                    *(float4*)&Xs[buf][r][cS] = make_float4(0.f, 0.f, 0.f, 0.f);
                }
            }
        };

        v8f acc0 = {}, acc1 = {}, acc2 = {}, acc3 = {};

        issue(0, 0);                           // prefetch first chunk
        for (int k0 = 0; k0 < DIM_IN; k0 += KT) {
            const int cur = (k0 >> 5) & 1;
            WAIT_ASYNC0();                     // my async loads for `cur` done
            __syncthreads();                   // everyone's loads visible; also
                                               // proves next buffer is reusable
            if (k0 + KT < DIM_IN) issue(cur ^ 1, k0 + KT);

            // Compute: K chunk in steps of 4 (V_WMMA_F32_16X16X4_F32)
            const float* xrow = &Xs[cur][wave * 16 + l16][hi];
            #pragma unroll
            for (int kk = 0; kk < KT; kk += 4) {
                const v2f a  = *(const v2f*)&xrow[kk];
                const v2f b0 = *(const v2f*)&Ws[cur][ 0 + l16][kk + hi];
                const v2f b1 = *(const v2f*)&Ws[cur][16 + l16][kk + hi];
                const v2f b2 = *(const v2f*)&Ws[cur][32 + l16][kk + hi];
                const v2f b3 = *(const v2f*)&Ws[cur][48 + l16][kk + hi];
                acc0 = __builtin_amdgcn_wmma_f32_16x16x4_f32(false, a, false, b0,
                                                             (short)0, acc0, false, false);
                acc1 = __builtin_amdgcn_wmma_f32_16x16x4_f32(false, a, false, b1,
                                                             (short)0, acc1, false, false);
                acc2 = __builtin_amdgcn_wmma_f32_16x16x4_f32(false, a, false, b2,
                                                             (short)0, acc2, false, false);
                acc3 = __builtin_amdgcn_wmma_f32_16x16x4_f32(false, a, false, b3,
                                                             (short)0, acc3, false, false);
            }
        }

        // Store: C/D layout -> VGPR v: lanes 0-15 M=v, lanes 16-31 M=v+8; N=lane&15
        const int mrow0 = wave * 16 + ((lane >> 4) << 3);
        #pragma unroll
        for (int v = 0; v < 8; ++v) {
            const int m = m0 + mrow0 + v;
            if (m < cnt) {
                const int tok = perm[base + m];
                float* orow = out + (size_t)tok * DIM_OUT + n0 + l16;
                orow[ 0] = acc0[v];
                orow[16] = acc1[v];
                orow[32] = acc2[v];
                orow[48] = acc3[v];
            }
        }
        __syncthreads();                       // block before next m0 reuses buf 0
    }
}

// ---------------- launch ----------------

extern "C" void kernel_launch(void* const* d_in, const int* in_sizes, int n_in,
                              void* d_out, int out_size, void* d_ws, size_t ws_size,
                              hipStream_t stream) {
    const float* inp    = (const float*)d_in[0];   // [T, IN]
    const int*   gate   = (const int*)d_in[1];     // [T]
    const float* weight = (const float*)d_in[2];   // [E, OUT, IN]
    float*       out    = (float*)d_out;           // [T, OUT]

    // workspace: counts[32] | offsets[32] | cursor[32] | perm[T]
    int* counts  = (int*)d_ws;
    int* offsets = counts + 32;
    int* cursor  = counts + 64;
    int* perm    = counts + 96;

    fmoe_zero_ws<<<1, 128, 0, stream>>>(counts);
    fmoe_histo<<<(NUM_T + 255) / 256, 256, 0, stream>>>(gate, counts);
    fmoe_scan<<<1, 32, 0, stream>>>(counts, offsets, cursor);
    fmoe_scatter<<<(NUM_T + 255) / 256, 256, 0, stream>>>(gate, cursor, perm);

    dim3 grid(DIM_OUT / NT, NUM_E);
    fmoe_gemm<<<grid, 256, 0, stream>>>(inp, weight, perm, counts, offsets, out);
}